// DRR_1236950582175
// MI455X (gfx1250) — compile-verified
//
#include <hip/hip_runtime.h>
#include <hip/hip_bf16.h>

#define VOLN   256
#define IMH    200
#define IMW    200
#define NCHUNK 8
#define PFAHEAD 6

typedef __attribute__((ext_vector_type(2))) float v2f;
typedef __attribute__((ext_vector_type(8))) float v8f;

__device__ __forceinline__ float lerpf(float a, float b, float f) {
    return fmaf(f, b - a, a);
}

__device__ __forceinline__ float vfetch(const float* __restrict__ vol, int x, int y, int z) {
    if ((unsigned)x < (unsigned)VOLN && (unsigned)y < (unsigned)VOLN && (unsigned)z < (unsigned)VOLN)
        return vol[(x * VOLN + y) * VOLN + z];
    return 0.0f;
}

__global__ __launch_bounds__(256) void drr_kernel(
    const float* __restrict__ vol,    // 256^3
    const float* __restrict__ kinv,   // B*9
    const float* __restrict__ rtinv,  // B*16
    const float* __restrict__ sddp,   // B
    const float* __restrict__ aff,    // 16
    const int*   __restrict__ nsp,    // 1
    float* __restrict__ out)          // B*H*W
{
    const int tx    = threadIdx.x;         // 0..31 -> image row v within tile
    const int ty    = threadIdx.y;         // 0..7  -> sample chunk
    const int vbase = blockIdx.x * 32;
    const int u     = blockIdx.y;          // image column
    const int b     = blockIdx.z;
    const int v     = vbase + tx;
    const int n     = *nsp;
    const float fn  = (float)(n - 1);
    const float inv_nm1 = 1.0f / fn;

    const float* K  = kinv  + 9  * b;
    const float* RT = rtinv + 16 * b;
    const float  sdd = sddp[b];

    // ---- per-pixel ray setup (exactly the reference math) ----
    const float pu = (float)u, pv = (float)v;
    const float tcx = fmaf(K[0], pu, fmaf(K[1], pv, K[2])) * sdd;
    const float tcy = fmaf(K[3], pu, fmaf(K[4], pv, K[5])) * sdd;
    const float tcz = fmaf(K[6], pu, fmaf(K[7], pv, K[8])) * sdd;

    const float rx = fmaf(RT[0], tcx, fmaf(RT[1], tcy, RT[2]  * tcz));
    const float ry = fmaf(RT[4], tcx, fmaf(RT[5], tcy, RT[6]  * tcz));
    const float rz = fmaf(RT[8], tcx, fmaf(RT[9], tcy, RT[10] * tcz));
    const float sx = RT[3], sy = RT[7], sz = RT[11];

    // vox(t) = basev + t*dirv  (affine folded in)
    const float bx = fmaf(aff[0], sx, fmaf(aff[1], sy, fmaf(aff[2],  sz, aff[3])));
    const float by = fmaf(aff[4], sx, fmaf(aff[5], sy, fmaf(aff[6],  sz, aff[7])));
    const float bz = fmaf(aff[8], sx, fmaf(aff[9], sy, fmaf(aff[10], sz, aff[11])));
    const float dx = fmaf(aff[0], rx, fmaf(aff[1], ry, aff[2]  * rz));
    const float dy = fmaf(aff[4], rx, fmaf(aff[5], ry, aff[6]  * rz));
    const float dz = fmaf(aff[8], rx, fmaf(aff[9], ry, aff[10] * rz));

    const float step = sqrtf(fmaf(rx, rx, fmaf(ry, ry, rz * rz))) * inv_nm1;

    // ---- conservative slab clip against the (-1, 256) support of the interpolant ----
    float t0 = 0.0f, t1 = 1.0f;
    {
        const float bs[3] = {bx, by, bz};
        const float ds[3] = {dx, dy, dz};
        #pragma unroll
        for (int i = 0; i < 3; ++i) {
            if (fabsf(ds[i]) > 1e-12f) {
                const float ta = (-1.0f - bs[i]) / ds[i];
                const float tb = ((float)VOLN - bs[i]) / ds[i];
                t0 = fmaxf(t0, fminf(ta, tb));
                t1 = fminf(t1, fmaxf(ta, tb));
            } else if (bs[i] <= -1.0f || bs[i] >= (float)VOLN) {
                t0 = 1.0f; t1 = -1.0f;   // empty
            }
        }
    }
    int slo = (int)fmaxf(0.0f, floorf(t0 * fn) - 1.0f);
    int shi = (int)fminf(fn,  ceilf(t1 * fn) + 1.0f);
    if (t1 < t0) { slo = 1; shi = 0; }

    // ---- this thread's chunk of samples ----
    const int csz = (n + NCHUNK - 1) / NCHUNK;
    int s  = max(slo, ty * csz);
    int se = min(shi, min(n - 1, (ty + 1) * csz - 1));

    const bool pix_valid = (v < IMH);

    float sum = 0.0f;
    if (pix_valid) {
        for (; s <= se; ++s) {
            const float t  = (float)s * inv_nm1;
            const float x  = fmaf(t, dx, bx);
            const float y  = fmaf(t, dy, by);
            const float z  = fmaf(t, dz, bz);
            const float xf = floorf(x), yf = floorf(y), zf = floorf(z);
            const int ix = (int)xf, iy = (int)yf, iz = (int)zf;
            const float fx = x - xf, fy = y - yf, fz = z - zf;

            float v000, v001, v010, v011, v100, v101, v110, v111;
            if ((unsigned)ix < (unsigned)(VOLN - 1) &&
                (unsigned)iy < (unsigned)(VOLN - 1) &&
                (unsigned)iz < (unsigned)(VOLN - 1)) {
                const float* p = vol + ((ix * VOLN + iy) * VOLN + iz);
                v000 = p[0];        v001 = p[1];
                v010 = p[VOLN];     v011 = p[VOLN + 1];
                v100 = p[VOLN*VOLN];          v101 = p[VOLN*VOLN + 1];
                v110 = p[VOLN*VOLN + VOLN];   v111 = p[VOLN*VOLN + VOLN + 1];
            } else {
                v000 = vfetch(vol, ix,   iy,   iz  );
                v001 = vfetch(vol, ix,   iy,   iz+1);
                v010 = vfetch(vol, ix,   iy+1, iz  );
                v011 = vfetch(vol, ix,   iy+1, iz+1);
                v100 = vfetch(vol, ix+1, iy,   iz  );
                v101 = vfetch(vol, ix+1, iy,   iz+1);
                v110 = vfetch(vol, ix+1, iy+1, iz  );
                v111 = vfetch(vol, ix+1, iy+1, iz+1);
            }
            const float c00 = lerpf(v000, v001, fz);
            const float c01 = lerpf(v010, v011, fz);
            const float c10 = lerpf(v100, v101, fz);
            const float c11 = lerpf(v110, v111, fz);
            const float c0  = lerpf(c00, c01, fy);
            const float c1  = lerpf(c10, c11, fy);
            sum += lerpf(c0, c1, fx);

            // prefetch a few samples ahead along the ray (first-touch HBM pass)
            const float tp = (float)(s + PFAHEAD) * inv_nm1;
            const float xp = fminf((float)(VOLN - 1), fmaxf(0.0f, fmaf(tp, dx, bx)));
            const float yp = fminf((float)(VOLN - 1), fmaxf(0.0f, fmaf(tp, dy, by)));
            const float zp = fminf((float)(VOLN - 1), fmaf(tp, dz, fmaxf(0.0f, bz)));
            const int ixp = (int)xp, iyp = (int)yp,
                      izp = (int)fminf((float)(VOLN - 1), fmaxf(0.0f, zp));
            __builtin_prefetch(vol + ((ixp * VOLN + iyp) * VOLN + izp), 0, 3);
        }
    }

    // ---- cross-chunk reduction ----
    __shared__ float part[NCHUNK][32];
    __shared__ float stepv[32];
    part[ty][tx] = sum;                 // zero for invalid pixels
    if (ty == 0) stepv[tx] = step;
    __syncthreads();

    float* outb = out + (size_t)b * IMH * IMW;

#if defined(__HIP_DEVICE_COMPILE__) && __has_builtin(__builtin_amdgcn_wmma_f32_16x16x4_f32)
    // Reduce 8 chunk-partials x 32 pixels with V_WMMA_F32_16X16X4_F32.
    // B = all-ones => D[m,n] = sum_k A[m,k] + C[m,n]  (B layout irrelevant).
    // A 16x4 f32 layout (ISA 7.12.2): lanes 0-15 hold {K0,K1}, lanes 16-31 hold {K2,K3}.
    // D 16x16 f32 layout: lane 0 -> D[0..7,0] in VGPR 0..7, lane 16 -> D[8..15,0].
    if (ty == 0) {                       // one full wave, EXEC all ones
        const int lane = tx;
        const int m    = lane & 15;
        const int kb   = (lane >> 4) * 2;   // 0 or 2
        const v2f ones = {1.0f, 1.0f};
        #pragma unroll
        for (int h = 0; h < 2; ++h) {       // pixel halves 0..15 / 16..31
            v8f acc = {};
            v2f a;
            a.x = part[kb + 0][h * 16 + m];
            a.y = part[kb + 1][h * 16 + m];
            acc = __builtin_amdgcn_wmma_f32_16x16x4_f32(false, a, false, ones,
                                                        (short)0, acc, false, false);
            a.x = part[kb + 4][h * 16 + m];
            a.y = part[kb + 5][h * 16 + m];
            acc = __builtin_amdgcn_wmma_f32_16x16x4_f32(false, a, false, ones,
                                                        (short)0, acc, false, false);
            if (lane == 0 || lane == 16) {
                const int mbase = h * 16 + (lane >> 4) * 8;
                #pragma unroll
                for (int r = 0; r < 8; ++r) {
                    const int pr = mbase + r;       // pixel index in tile
                    const int vr = vbase + pr;      // image row
                    if (vr < IMH)
                        outb[vr * IMW + u] = acc[r] * stepv[pr];
                }
            }
        }
    }
#else
    if (ty == 0 && pix_valid) {
        float tot = 0.0f;
        #pragma unroll
        for (int c = 0; c < NCHUNK; ++c) tot += part[c][tx];
        outb[v * IMW + u] = tot * step;
    }
#endif
}

extern "C" void kernel_launch(void* const* d_in, const int* in_sizes, int n_in,
                              void* d_out, int out_size, void* d_ws, size_t ws_size,
                              hipStream_t stream) {
    const float* vol   = (const float*)d_in[0];
    const float* kinv  = (const float*)d_in[1];
    const float* rtinv = (const float*)d_in[2];
    const float* sdd   = (const float*)d_in[3];
    const float* aff   = (const float*)d_in[4];
    const int*   nsp   = (const int*)d_in[5];
    float* out = (float*)d_out;

    int B = in_sizes[1] / 9;
    if (B < 1) B = 1;

    dim3 block(32, NCHUNK);
    dim3 grid((IMH + 31) / 32, IMW, B);
    drr_kernel<<<grid, block, 0, stream>>>(vol, kinv, rtinv, sdd, aff, nsp, out);
}